// GCN_72713796321514
// MI455X (gfx1250) — compile-verified
//
#include <hip/hip_runtime.h>
#include <hip/hip_bf16.h>

// ---------------- problem constants ----------------
#define NN   400                  // nodes
#define BB   64                   // batch
#define NTRI (NN*(NN-1)/2)        // 79800
#define BN_SAMP (BB*NN)           // 25600
#define EPSV 1e-5f

typedef float v2f __attribute__((ext_vector_type(2)));
typedef float v4f __attribute__((ext_vector_type(4)));
typedef float v8f __attribute__((ext_vector_type(8)));

// fp32 WMMA: D(16x16) = A(16x4) * B(4x16) + C   (8-arg VOP3P form)
__device__ __forceinline__ v8f wmma4(v2f a, v2f b, v8f c) {
  return __builtin_amdgcn_wmma_f32_16x16x4_f32(false, a, false, b, (short)0, c,
                                               false, false);
}

__device__ __forceinline__ int triIdx(int i, int j) { // requires i < j
  return i * (2 * NN - i - 1) / 2 + (j - i - 1);
}

// ---------------- adjacency construction ----------------

// one wave per (b,i) row: degree -> dinv = rsqrt(1 + #nonzero offdiag)
__global__ void degree_kernel(const float* __restrict__ data,
                              float* __restrict__ dinv) {
  int wave = (blockIdx.x * blockDim.x + threadIdx.x) >> 5;
  int lane = threadIdx.x & 31;
  if (wave >= BB * NN) return;
  int b = wave / NN, i = wave % NN;
  const float* dp = data + (size_t)b * NTRI;
  float cnt = 0.f;
  for (int j = lane; j < NN; j += 32) {
    if (j == i) continue;
    int ii = (i < j) ? i : j, jj = (i < j) ? j : i;
    cnt += (dp[triIdx(ii, jj)] != 0.f) ? 1.f : 0.f;
  }
  for (int off = 16; off > 0; off >>= 1) cnt += __shfl_down(cnt, off, 32);
  if (lane == 0) dinv[wave] = rsqrtf(1.f + cnt);
}

// dense normalized adjacency an[b][i][j] = a_ij * dinv_i * dinv_j
__global__ void adj_kernel(const float* __restrict__ data,
                           const float* __restrict__ dinv,
                           float* __restrict__ an) {
  size_t idx = (size_t)blockIdx.x * blockDim.x + threadIdx.x;
  if (idx >= (size_t)BB * NN * NN) return;
  int b = (int)(idx / (NN * NN));
  int rem = (int)(idx % (NN * NN));
  int i = rem / NN, j = rem % NN;
  float a;
  if (i == j) a = 1.f;
  else {
    int ii = (i < j) ? i : j, jj = (i < j) ? j : i;
    a = (data[(size_t)b * NTRI + triIdx(ii, jj)] != 0.f) ? 1.f : 0.f;
  }
  an[idx] = a * dinv[b * NN + i] * dinv[b * NN + j];
}

// one wave per row: trow[b,i] = sum_j an[b,i,j]
__global__ void rowsum_kernel(const float* __restrict__ an,
                              float* __restrict__ trow) {
  int wave = (blockIdx.x * blockDim.x + threadIdx.x) >> 5;
  int lane = threadIdx.x & 31;
  if (wave >= BB * NN) return;
  const float* rp = an + (size_t)wave * NN;
  float s = 0.f;
  for (int j = lane; j < NN; j += 32) s += rp[j];
  for (int off = 16; off > 0; off >>= 1) s += __shfl_down(s, off, 32);
  if (lane == 0) trow[wave] = s;
}

// ---------------- layer 0 (x = ones collapses to a rank-1 update) ----------
__global__ void layer0_kernel(const float* __restrict__ trow,
                              const float* __restrict__ w0,   // [64,1]
                              const float* __restrict__ b0,   // [64]
                              float* __restrict__ y) {        // [25600,64]
  int idx = blockIdx.x * blockDim.x + threadIdx.x;
  if (idx >= BN_SAMP * 64) return;
  int s = idx >> 6, f = idx & 63;
  float v = trow[s] * w0[f] + b0[f];
  y[idx] = v > 0.f ? v : 0.f;
}

// ---------------- batchnorm ----------------
__global__ void bn_reduce_kernel(const float* __restrict__ y, int dout,
                                 float* __restrict__ stats) { // [mean|invstd]
  int f = blockIdx.x, tid = threadIdx.x;
  float s = 0.f, s2 = 0.f;
  for (int r = tid; r < BN_SAMP; r += 256) {
    float v = y[(size_t)r * dout + f];
    s += v; s2 += v * v;
  }
  __shared__ float ss[256], ss2[256];
  ss[tid] = s; ss2[tid] = s2; __syncthreads();
  for (int o = 128; o > 0; o >>= 1) {
    if (tid < o) { ss[tid] += ss[tid + o]; ss2[tid] += ss2[tid + o]; }
    __syncthreads();
  }
  if (tid == 0) {
    float m = ss[0] * (1.f / BN_SAMP);
    float var = ss2[0] * (1.f / BN_SAMP) - m * m;
    stats[f] = m;
    stats[dout + f] = rsqrtf(var + EPSV);
  }
}

__global__ void bn_norm_kernel(const float* __restrict__ y, int dout,
                               const float* __restrict__ stats,
                               const float* __restrict__ g,
                               const float* __restrict__ beta,
                               float* __restrict__ x) {
  size_t idx = (size_t)blockIdx.x * blockDim.x + threadIdx.x;
  if (idx >= (size_t)BN_SAMP * dout) return;
  int f = (int)(idx % dout);
  x[idx] = (y[idx] - stats[f]) * stats[dout + f] * g[f] + beta[f];
}

// ---------------- Ht = (X * W^T)^T per batch -----------------------------
// one wave per 16x32 C-tile; LDS transpose epilogue so angemm's B-matrix
// (Ht rows) can be loaded as contiguous b64 pairs.
// Ht layout: [B][dout][400]
__global__ void hgemm_kernel(const float* __restrict__ X, int din,
                             const float* __restrict__ W, int dout,
                             float* __restrict__ Ht) {
  __shared__ float lds[32 * 17];     // [col][row], padded stride 17
  int m0 = blockIdx.x * 16;          // global row (b*400 + i), 16|400 => one batch
  int j0 = blockIdx.y * 32;          // output feature tile
  int lane = threadIdx.x;
  int half = lane >> 4, mn = lane & 15;
  v8f acc0 = {}; v8f acc1 = {};
  const float* arow  = X + (size_t)(m0 + mn) * din;
  const float* brow0 = W + (size_t)(j0 + mn) * din;
  const float* brow1 = W + (size_t)(j0 + 16 + mn) * din;
  for (int k0 = 0; k0 < din; k0 += 4) {
    int kk = k0 + half * 2;
    v2f a  = *(const v2f*)(arow + kk);
    v2f b0 = *(const v2f*)(brow0 + kk);
    v2f b1 = *(const v2f*)(brow1 + kk);
    acc0 = wmma4(a, b0, acc0);
    acc1 = wmma4(a, b1, acc1);
  }
  for (int r = 0; r < 8; ++r) {
    lds[mn * 17 + r + 8 * half]        = acc0[r];
    lds[(mn + 16) * 17 + r + 8 * half] = acc1[r];
  }
  __syncthreads();
  int b = m0 / NN, il = m0 % NN;
  float* outp = Ht + ((size_t)b * dout + j0 + lane) * NN + il;
  for (int c = 0; c < 4; ++c) {
    v4f v;
    v.x = lds[lane * 17 + 4 * c + 0];
    v.y = lds[lane * 17 + 4 * c + 1];
    v.z = lds[lane * 17 + 4 * c + 2];
    v.w = lds[lane * 17 + 4 * c + 3];
    *(v4f*)(outp + 4 * c) = v;
  }
}

// ---------------- Y = AN * H + bias, relu (batched) ----------------------
// grid (25, dout/64, B); one wave per 16x64 tile; A fragment reused x4.
// All loads are contiguous b64 (A rows of an, B rows of Ht).
__global__ void angemm_kernel(const float* __restrict__ an,
                              const float* __restrict__ Ht, int dout,
                              const float* __restrict__ bias,
                              float* __restrict__ Y) {
  int bb = blockIdx.z;
  int i0 = blockIdx.x * 16;
  int j0 = blockIdx.y * 64;
  int lane = threadIdx.x;
  int half = lane >> 4, mn = lane & 15;
  const float* A   = an + (size_t)bb * NN * NN;
  const float* Htb = Ht + (size_t)bb * dout * NN;
  v8f acc0 = {}, acc1 = {}, acc2 = {}, acc3 = {};
  const float* arow = A + (size_t)(i0 + mn) * NN;
  const float* br0 = Htb + (size_t)(j0 + mn) * NN;
  const float* br1 = Htb + (size_t)(j0 + 16 + mn) * NN;
  const float* br2 = Htb + (size_t)(j0 + 32 + mn) * NN;
  const float* br3 = Htb + (size_t)(j0 + 48 + mn) * NN;
  for (int k0 = 0; k0 < NN; k0 += 4) {
    int kk = k0 + half * 2;
    if (k0 + 64 < NN) {
      __builtin_prefetch(arow + kk + 64, 0, 0);
      __builtin_prefetch(br0 + kk + 64, 0, 0);
    }
    v2f a = *(const v2f*)(arow + kk);
    acc0 = wmma4(a, *(const v2f*)(br0 + kk), acc0);
    acc1 = wmma4(a, *(const v2f*)(br1 + kk), acc1);
    acc2 = wmma4(a, *(const v2f*)(br2 + kk), acc2);
    acc3 = wmma4(a, *(const v2f*)(br3 + kk), acc3);
  }
  v8f accs[4] = {acc0, acc1, acc2, acc3};
  for (int f = 0; f < 4; ++f) {
    float bv = bias[j0 + 16 * f + mn];
    for (int r = 0; r < 8; ++r) {
      int row = i0 + r + 8 * half;
      float y = accs[f][r] + bv;
      y = y > 0.f ? y : 0.f;
      Y[((size_t)bb * NN + row) * dout + j0 + 16 * f + mn] = y;
    }
  }
}

// ---------------- pooling FC: [64 x 102400] @ [102400 x 256] -------------
__global__ void zero_kernel(float* __restrict__ p, int n) {
  int i = blockIdx.x * blockDim.x + threadIdx.x;
  if (i < n) p[i] = 0.f;
}

#define POOL_K (NN * 256)     // 102400
#define KSPLIT 64
// grid (4, 4, KSPLIT); one wave per 16x64 tile-chunk, atomic K reduction
__global__ void poolgemm_kernel(const float* __restrict__ X,
                                const float* __restrict__ PW,
                                float* __restrict__ acc) {
  int m0 = blockIdx.x * 16;
  int j0 = blockIdx.y * 64;
  int kbase = blockIdx.z * (POOL_K / KSPLIT);
  int lane = threadIdx.x;
  int half = lane >> 4, mn = lane & 15;
  const float* arow = X + (size_t)(m0 + mn) * POOL_K;
  const float* br0 = PW + (size_t)(j0 + mn) * POOL_K;
  const float* br1 = PW + (size_t)(j0 + 16 + mn) * POOL_K;
  const float* br2 = PW + (size_t)(j0 + 32 + mn) * POOL_K;
  const float* br3 = PW + (size_t)(j0 + 48 + mn) * POOL_K;
  v8f acc0 = {}, acc1 = {}, acc2 = {}, acc3 = {};
  int kend = kbase + POOL_K / KSPLIT;
  for (int k0 = kbase; k0 < kend; k0 += 4) {
    int kk = k0 + half * 2;
    if (k0 + 128 < kend) {
      __builtin_prefetch(arow + kk + 128, 0, 0);
      __builtin_prefetch(br0 + kk + 128, 0, 0);
    }
    v2f a = *(const v2f*)(arow + kk);
    acc0 = wmma4(a, *(const v2f*)(br0 + kk), acc0);
    acc1 = wmma4(a, *(const v2f*)(br1 + kk), acc1);
    acc2 = wmma4(a, *(const v2f*)(br2 + kk), acc2);
    acc3 = wmma4(a, *(const v2f*)(br3 + kk), acc3);
  }
  v8f accs[4] = {acc0, acc1, acc2, acc3};
  for (int f = 0; f < 4; ++f) {
    for (int r = 0; r < 8; ++r) {
      int row = m0 + r + 8 * half;
      __hip_atomic_fetch_add(&acc[row * 256 + j0 + 16 * f + mn], accs[f][r],
                             __ATOMIC_RELAXED, __HIP_MEMORY_SCOPE_AGENT);
    }
  }
}

__global__ void pool_epilogue_kernel(const float* __restrict__ acc,
                                     const float* __restrict__ pb,
                                     float* __restrict__ out) {
  int idx = blockIdx.x * blockDim.x + threadIdx.x;
  if (idx >= BB * 256) return;
  float v = acc[idx] + pb[idx & 255];
  out[idx] = v > 0.f ? v : 0.f;
}

// ---------------- launch ----------------
extern "C" void kernel_launch(void* const* d_in, const int* in_sizes, int n_in,
                              void* d_out, int out_size, void* d_ws, size_t ws_size,
                              hipStream_t stream) {
  const float* data = (const float*)d_in[0];
  const float* w[3]  = { (const float*)d_in[1], (const float*)d_in[5], (const float*)d_in[9]  };
  const float* cb[3] = { (const float*)d_in[2], (const float*)d_in[6], (const float*)d_in[10] };
  const float* g[3]  = { (const float*)d_in[3], (const float*)d_in[7], (const float*)d_in[11] };
  const float* bt[3] = { (const float*)d_in[4], (const float*)d_in[8], (const float*)d_in[12] };
  const float* pw = (const float*)d_in[13];
  const float* pb = (const float*)d_in[14];
  float* out = (float*)d_out;

  float* ws = (float*)d_ws;
  size_t off = 0;
  float* an    = ws + off; off += (size_t)BB * NN * NN;   // 10.24 M
  float* dinv  = ws + off; off += BB * NN;
  float* trow  = ws + off; off += BB * NN;
  float* xbuf  = ws + off; off += (size_t)BB * NN * 256;
  float* htbuf = ws + off; off += (size_t)BB * NN * 256;  // transposed H
  float* ybuf  = ws + off; off += (size_t)BB * NN * 256;
  float* stats = ws + off; off += 512;
  float* pacc  = ws + off; off += BB * 256;

  const int dims[4] = {1, 64, 128, 256};

  // adjacency
  degree_kernel<<<(BB * NN) / 8, 256, 0, stream>>>(data, dinv);
  adj_kernel<<<(int)(((size_t)BB * NN * NN) / 256), 256, 0, stream>>>(data, dinv, an);
  rowsum_kernel<<<(BB * NN) / 8, 256, 0, stream>>>(an, trow);

  // layer 0 (input is all-ones -> rank-1)
  layer0_kernel<<<(BN_SAMP * 64) / 256, 256, 0, stream>>>(trow, w[0], cb[0], ybuf);
  bn_reduce_kernel<<<64, 256, 0, stream>>>(ybuf, 64, stats);
  bn_norm_kernel<<<(BN_SAMP * 64) / 256, 256, 0, stream>>>(ybuf, 64, stats, g[0], bt[0], xbuf);

  // layers 1..2
  for (int L = 1; L <= 2; ++L) {
    int din = dims[L], dout = dims[L + 1];
    hgemm_kernel<<<dim3(BN_SAMP / 16, dout / 32), 32, 0, stream>>>(xbuf, din, w[L], dout, htbuf);
    angemm_kernel<<<dim3(NN / 16, dout / 64, BB), 32, 0, stream>>>(an, htbuf, dout, cb[L], ybuf);
    bn_reduce_kernel<<<dout, 256, 0, stream>>>(ybuf, dout, stats);
    bn_norm_kernel<<<(int)(((size_t)BN_SAMP * dout) / 256), 256, 0, stream>>>(
        ybuf, dout, stats, g[L], bt[L], xbuf);
  }

  // pooling FC
  zero_kernel<<<(BB * 256) / 256, 256, 0, stream>>>(pacc, BB * 256);
  poolgemm_kernel<<<dim3(BB / 16, 256 / 64, KSPLIT), 32, 0, stream>>>(xbuf, pw, pacc);
  pool_epilogue_kernel<<<(BB * 256) / 256, 256, 0, stream>>>(pacc, pb, out);
}